// LinkPredict_21809843929178
// MI455X (gfx1250) — compile-verified
//
#include <hip/hip_runtime.h>

typedef __attribute__((ext_vector_type(2))) float v2f;
typedef __attribute__((ext_vector_type(8))) float v8f;

#define N_NODES 100000
#define H_DIM   128
#define N_RELS  90
#define N_BASES 4
#define N_EDGES 800000
#define XB_COLS (N_BASES * H_DIM)   // 512

// ---------------------------------------------------------------------------
// Kernel 1: zero the accumulator (d_out) and the per-node counter (in ws).
// ---------------------------------------------------------------------------
__global__ __launch_bounds__(256) void rgcn_zero(float* __restrict__ out,
                                                 float* __restrict__ cnt) {
    const int idx = blockIdx.x * 256 + threadIdx.x;   // 12.8M threads
    if (idx < N_NODES * H_DIM) out[idx] = 0.0f;
    if (idx < N_NODES)         cnt[idx] = 0.0f;
}

// ---------------------------------------------------------------------------
// Kernel 2: basis projection GEMM via V_WMMA_F32_16X16X4_F32.
//   xb[n][b*128+o] = sum_i emb[h[n]][i] * weight[b][i][o]
// One wave per 16x16 tile of C ([100000 x 512]); K=128 in 32 steps of 4.
// A 16x4 f32 layout: lane<16 -> {K=0,K=1}, lane>=16 -> {K=2,K=3}, M = lane&15.
// B 4x16 f32 layout (mirror): lane<16 -> rows {0,1}, lane>=16 -> rows {2,3},
// N = lane&15. C/D: VGPR v -> M = v + 8*(lane>=16), N = lane&15.
// ---------------------------------------------------------------------------
__global__ __launch_bounds__(128) void rgcn_basis_gemm(
    const int*   __restrict__ h_ids,    // [N_NODES]
    const float* __restrict__ emb,      // [N_NODES][128]
    const float* __restrict__ weight,   // [4][128][128]  (b, i, o)
    float*       __restrict__ xb)       // [N_NODES][512]
{
    const int wave = threadIdx.x >> 5;
    const int lane = threadIdx.x & 31;
    const int tile = blockIdx.x * 4 + wave;   // 200000 tiles total
    const int mt   = tile >> 5;               // 0..6249
    const int nt   = tile & 31;               // 0..31
    const int m0   = mt * 16;
    const int n0   = nt * 16;                 // column in [0,512)
    const int half = lane >> 4;               // 0 or 1
    const int l15  = lane & 15;

    // A: this lane's source row (through the h gather, per reference)
    const int    node = h_ids[m0 + l15];
    const float* arow = emb + (size_t)node * H_DIM;

    // B: this lane's output column. All 16 cols of a tile share one basis
    // because n0 is a multiple of 16 and 128 is a multiple of 16.
    const int    col   = n0 + l15;
    const int    basis = col >> 7;
    const int    oo    = col & 127;
    const float* bcol  = weight + (size_t)basis * H_DIM * H_DIM + oo; // stride H_DIM over i

    v8f c = {0.f, 0.f, 0.f, 0.f, 0.f, 0.f, 0.f, 0.f};
    #pragma unroll 8
    for (int k0 = 0; k0 < H_DIM; k0 += 4) {
        const int ka = k0 + half * 2;
        v2f a, b;
        a.x = arow[ka];
        a.y = arow[ka + 1];
        b.x = bcol[(size_t)ka * H_DIM];
        b.y = bcol[(size_t)(ka + 1) * H_DIM];
        // D = A*B + C, full f32 accumulate on the matrix pipe
        c = __builtin_amdgcn_wmma_f32_16x16x4_f32(
                /*neg_a=*/false, a, /*neg_b=*/false, b,
                /*c_mod=*/(short)0, c, /*reuse_a=*/false, /*reuse_b=*/false);
    }

    // Store the 16x16 tile: VGPR v holds row m0 + v + 8*half, col n0 + l15.
    float* outp = xb + (size_t)(m0 + half * 8) * XB_COLS + n0 + l15;
    #pragma unroll
    for (int v = 0; v < 8; ++v)
        outp[(size_t)v * XB_COLS] = c[v];
}

// ---------------------------------------------------------------------------
// Kernel 3: per-edge message + scatter. One wave (32 lanes) per edge; each
// lane owns 4 contiguous outputs (float4 gathers -> global_load_b128).
// ---------------------------------------------------------------------------
__global__ __launch_bounds__(256) void rgcn_edge_scatter(
    const int*   __restrict__ r,        // [E]
    const float* __restrict__ norm,     // [E]
    const int*   __restrict__ src,      // [E]
    const int*   __restrict__ dst,      // [E]
    const float* __restrict__ w_comp,   // [90][4]
    const float* __restrict__ xb,       // [N][512]
    float*       __restrict__ agg,      // d_out [N][128] (accumulator)
    float*       __restrict__ cnt)      // [N]
{
    const int e = blockIdx.x * 8 + (threadIdx.x >> 5);
    if (e >= N_EDGES) return;
    const int lane = threadIdx.x & 31;

    const int   rel = r[e];
    const float nrm = norm[e];
    const int   s   = src[e];
    const int   d   = dst[e];

    const float c0 = w_comp[rel * 4 + 0];
    const float c1 = w_comp[rel * 4 + 1];
    const float c2 = w_comp[rel * 4 + 2];
    const float c3 = w_comp[rel * 4 + 3];

    // xb row: 512 floats = 4 bases x 128; lane covers o = 4*lane .. 4*lane+3
    const float4* p = (const float4*)(xb + (size_t)s * XB_COLS) + lane;
    const float4 v0 = p[0];    // basis 0
    const float4 v1 = p[32];   // basis 1 (+128 floats)
    const float4 v2 = p[64];   // basis 2
    const float4 v3 = p[96];   // basis 3

    float4 m;
    m.x = nrm * (c0 * v0.x + c1 * v1.x + c2 * v2.x + c3 * v3.x);
    m.y = nrm * (c0 * v0.y + c1 * v1.y + c2 * v2.y + c3 * v3.y);
    m.z = nrm * (c0 * v0.z + c1 * v1.z + c2 * v2.z + c3 * v3.z);
    m.w = nrm * (c0 * v0.w + c1 * v1.w + c2 * v2.w + c3 * v3.w);

    float* o = agg + (size_t)d * H_DIM + lane * 4;
    unsafeAtomicAdd(o + 0, m.x);   // -> global_atomic_add_f32 (L2 atomic units)
    unsafeAtomicAdd(o + 1, m.y);
    unsafeAtomicAdd(o + 2, m.z);
    unsafeAtomicAdd(o + 3, m.w);
    if (lane == 0) unsafeAtomicAdd(cnt + d, 1.0f);
}

// ---------------------------------------------------------------------------
// Kernel 4: mean over destinations + bias.
// ---------------------------------------------------------------------------
__global__ __launch_bounds__(256) void rgcn_finalize(
    float*       __restrict__ out,
    const float* __restrict__ cnt,
    const float* __restrict__ bias)
{
    const int idx = blockIdx.x * 256 + threadIdx.x;
    if (idx >= N_NODES * H_DIM) return;
    const int n = idx >> 7;
    const int o = idx & 127;
    const float c = cnt[n];
    out[idx] = out[idx] / fmaxf(c, 1.0f) + bias[o];
}

// ---------------------------------------------------------------------------
extern "C" void kernel_launch(void* const* d_in, const int* in_sizes, int n_in,
                              void* d_out, int out_size, void* d_ws, size_t ws_size,
                              hipStream_t stream) {
    const int*   h_ids  = (const int*)  d_in[0];
    const int*   r      = (const int*)  d_in[1];
    const float* norm   = (const float*)d_in[2];
    const int*   src    = (const int*)  d_in[3];
    const int*   dst    = (const int*)  d_in[4];
    const float* emb    = (const float*)d_in[5];
    const float* weight = (const float*)d_in[6];
    const float* w_comp = (const float*)d_in[7];
    const float* bias   = (const float*)d_in[8];

    float* out = (float*)d_out;
    float* xb  = (float*)d_ws;                       // 100000*512 f32 = 204.8 MB
    float* cnt = xb + (size_t)N_NODES * XB_COLS;     // +400 KB

    // 12.8M elems / 256
    rgcn_zero        <<<50000, 256, 0, stream>>>(out, cnt);
    // 6250 M-tiles * 32 N-tiles = 200000 waves, 4 waves/block
    rgcn_basis_gemm  <<<50000, 128, 0, stream>>>(h_ids, emb, weight, xb);
    // 800000 edges, 8 edges (waves) per block
    rgcn_edge_scatter<<<100000, 256, 0, stream>>>(r, norm, src, dst, w_comp, xb, out, cnt);
    rgcn_finalize    <<<50000, 256, 0, stream>>>(out, cnt, bias);
}